// GINBackbone_10943576670412
// MI455X (gfx1250) — compile-verified
//
#include <hip/hip_runtime.h>
#include <hip/hip_bf16.h>
#include <math.h>

#define N_NODES 100000
#define N_EDGES 1600000
#define DIM 256
#define N_LAYERS 4
#define LN_EPS 1e-5f

#define TM 64            // nodes per workgroup tile
#define LDS_STRIDE 260   // 256 + 4 pad -> 16 consecutive rows hit distinct LDS banks

typedef __attribute__((ext_vector_type(2))) float v2f;
typedef __attribute__((ext_vector_type(8))) float v8f;

// ---------------------------------------------------------------------------
// Weight pre-pack: W (row-major K x N) -> Wp[kq][half][n] float2 =
//   ( W[4*kq + 2*half][n], W[4*kq + 2*half + 1][n] )
// so a B-matrix WMMA fragment is a single coalesced b64 load per lane.
// Handles nmat stacked DIM x DIM matrices.
// ---------------------------------------------------------------------------
__global__ __launch_bounds__(256) void pack_weights(
    const float* __restrict__ W, float2* __restrict__ Wp, int nmat) {
  int idx = blockIdx.x * 256 + threadIdx.x;  // over nmat * (DIM/4) * 2 * DIM
  int total = nmat * (DIM / 4) * 2 * DIM;
  if (idx >= total) return;
  int n = idx % DIM;
  int t = idx / DIM;
  int h = t % 2;
  int t2 = t / 2;
  int kq = t2 % (DIM / 4);
  int m = t2 / (DIM / 4);
  const float* Wm = W + (size_t)m * DIM * DIM;
  float2 v;
  v.x = Wm[(size_t)(4 * kq + 2 * h) * DIM + n];
  v.y = Wm[(size_t)(4 * kq + 2 * h + 1) * DIM + n];
  Wp[idx] = v;
}

// ---------------------------------------------------------------------------
// Edge scatter: agg[dst] += h[src]   (sum aggregation)
// One wave32 per edge: readfirstlane makes src/dst scalar (SMEM) loads;
// each lane gathers 8 contiguous floats (2x b128) and issues 8 f32 atomics.
// Grid divides exactly (1.6M edges / 8 per block) -> no bounds check.
// ---------------------------------------------------------------------------
__global__ __launch_bounds__(256) void gin_scatter(
    const float* __restrict__ h, const int* __restrict__ src,
    const int* __restrict__ dst, float* __restrict__ agg) {
  int edge = __builtin_amdgcn_readfirstlane(blockIdx.x * 8 + (threadIdx.x >> 5));
  int lane = threadIdx.x & 31;
  int s = src[edge];
  int d = dst[edge];
  const float4* hp = (const float4*)(h + (size_t)s * DIM);
  float4 v0 = hp[lane * 2];
  float4 v1 = hp[lane * 2 + 1];
  float* o = agg + (size_t)d * DIM + lane * 8;
  atomicAdd(o + 0, v0.x);
  atomicAdd(o + 1, v0.y);
  atomicAdd(o + 2, v0.z);
  atomicAdd(o + 3, v0.w);
  atomicAdd(o + 4, v1.x);
  atomicAdd(o + 5, v1.y);
  atomicAdd(o + 6, v1.z);
  atomicAdd(o + 7, v1.w);
}

// ---------------------------------------------------------------------------
// Fused node update for one layer:
//   x  = (1+eps)*h + agg
//   y  = relu(x @ W1 + b1)
//   z  = y @ W2 + b2
//   o  = relu(LN(z)) + h
// 256 threads = 8 wave32. 64-node tile. WMMA f32 16x16x4, K=256 in steps of 4.
// Wave w: M-tile mt = w/2 (rows mt*16..+15), N-tiles (w&1)*8 .. +7; the A
// fragment is loaded once per k-quad from LDS and reused across 8 WMMAs.
// ---------------------------------------------------------------------------
__global__ __launch_bounds__(256) void gin_mlp(
    const float* __restrict__ h, const float* __restrict__ agg,
    const v2f* __restrict__ Wp1, const float* __restrict__ b1,
    const v2f* __restrict__ Wp2, const float* __restrict__ b2,
    const float* __restrict__ eps, int layer,
    const float* __restrict__ gamma, const float* __restrict__ beta,
    float* __restrict__ out) {
  __shared__ float lx[TM * LDS_STRIDE];  // x tile, later reused for z tile
  __shared__ float ly[TM * LDS_STRIDE];  // y tile (hidden activations)

  const int node0 = blockIdx.x * TM;
  const int tid = threadIdx.x;
  const float e1 = 1.0f + eps[layer];

  // ---- Phase 1: stage x = (1+eps)*h + agg into LDS (zeros past N_NODES) ----
  for (int i = tid; i < TM * (DIM / 4); i += 256) {
    int r = i / (DIM / 4);
    int c = (i % (DIM / 4)) * 4;
    int node = node0 + r;
    float4 v;
    if (node < N_NODES) {
      float4 hv = *(const float4*)(h + (size_t)node * DIM + c);
      float4 av = *(const float4*)(agg + (size_t)node * DIM + c);
      v.x = e1 * hv.x + av.x;
      v.y = e1 * hv.y + av.y;
      v.z = e1 * hv.z + av.z;
      v.w = e1 * hv.w + av.w;
    } else {
      v.x = v.y = v.z = v.w = 0.0f;
    }
    *(float4*)(&lx[r * LDS_STRIDE + c]) = v;  // stride%4==0 -> 16B aligned
  }
  __syncthreads();

  const int wave = tid >> 5;
  const int lane = tid & 31;
  const int half = lane >> 4;  // 0: lanes 0-15 (K=k0,k0+1), 1: lanes 16-31 (K=k0+2,k0+3)
  const int l16 = lane & 15;
  const int mt = wave >> 1;           // 0..3
  const int ntBase = (wave & 1) * 8;  // 0 or 8
  const int mrow = mt * 16 + l16;     // A-matrix row this lane supplies

  // ---- Phase 2: GEMM1  y = relu(x @ W1 + b1) ----
  {
    v8f acc[8] = {};
#pragma unroll 4
    for (int k0 = 0; k0 < DIM; k0 += 4) {
      v2f a;
      a.x = lx[mrow * LDS_STRIDE + k0 + 2 * half];
      a.y = lx[mrow * LDS_STRIDE + k0 + 2 * half + 1];
      const v2f* wrow = Wp1 + (size_t)((k0 >> 2) * 2 + half) * DIM;
#pragma unroll
      for (int i = 0; i < 8; ++i) {
        int n = (ntBase + i) * 16 + l16;
        v2f b = wrow[n];  // packed pair of K-rows: one b64 load, coalesced
        acc[i] = __builtin_amdgcn_wmma_f32_16x16x4_f32(
            false, a, false, b, (short)0, acc[i], false, false);
      }
    }
    // bias + relu, D layout: VGPR r -> M = mt*16 + r + 8*half, N = tile*16 + l16
#pragma unroll
    for (int i = 0; i < 8; ++i) {
      int n = (ntBase + i) * 16 + l16;
      float bias = b1[n];
#pragma unroll
      for (int r = 0; r < 8; ++r) {
        int m = mt * 16 + r + 8 * half;
        float v = acc[i][r] + bias;
        ly[m * LDS_STRIDE + n] = v > 0.0f ? v : 0.0f;
      }
    }
  }
  __syncthreads();

  // ---- Phase 3: GEMM2  z = y @ W2 + b2  (z written back into lx) ----
  {
    v8f acc[8] = {};
#pragma unroll 4
    for (int k0 = 0; k0 < DIM; k0 += 4) {
      v2f a;
      a.x = ly[mrow * LDS_STRIDE + k0 + 2 * half];
      a.y = ly[mrow * LDS_STRIDE + k0 + 2 * half + 1];
      const v2f* wrow = Wp2 + (size_t)((k0 >> 2) * 2 + half) * DIM;
#pragma unroll
      for (int i = 0; i < 8; ++i) {
        int n = (ntBase + i) * 16 + l16;
        v2f b = wrow[n];
        acc[i] = __builtin_amdgcn_wmma_f32_16x16x4_f32(
            false, a, false, b, (short)0, acc[i], false, false);
      }
    }
#pragma unroll
    for (int i = 0; i < 8; ++i) {
      int n = (ntBase + i) * 16 + l16;
      float bias = b2[n];
#pragma unroll
      for (int r = 0; r < 8; ++r) {
        int m = mt * 16 + r + 8 * half;
        lx[m * LDS_STRIDE + n] = acc[i][r] + bias;  // lx free after GEMM1
      }
    }
  }
  __syncthreads();

  // ---- Phase 4: LayerNorm + ReLU + residual; each wave owns 8 rows ----
  for (int rr = wave; rr < TM; rr += 8) {
    int node = node0 + rr;
    float s = 0.0f, sq = 0.0f;
    float vals[8];
#pragma unroll
    for (int j = 0; j < 8; ++j) {
      float v = lx[rr * LDS_STRIDE + lane + 32 * j];
      vals[j] = v;
      s += v;
      sq += v * v;
    }
#pragma unroll
    for (int off = 16; off >= 1; off >>= 1) {  // wave32 butterfly reduction
      s += __shfl_xor(s, off, 32);
      sq += __shfl_xor(sq, off, 32);
    }
    float mu = s * (1.0f / DIM);
    float var = sq * (1.0f / DIM) - mu * mu;
    float rs = rsqrtf(var + LN_EPS);
    if (node < N_NODES) {
#pragma unroll
      for (int j = 0; j < 8; ++j) {
        int c = lane + 32 * j;
        float v = (vals[j] - mu) * rs * gamma[c] + beta[c];
        v = v > 0.0f ? v : 0.0f;
        out[(size_t)node * DIM + c] = v + h[(size_t)node * DIM + c];
      }
    }
  }
}

// ---------------------------------------------------------------------------
// Host orchestration: pack weights once, then 4 layers of
// memset(agg) -> scatter -> fused MLP.  Ping-pong: h -> bufA -> d_out -> ...
// Workspace: [agg][bufA][Wp1: 4x64KiB*4][Wp2]  (~207 MB)
// ---------------------------------------------------------------------------
extern "C" void kernel_launch(void* const* d_in, const int* in_sizes, int n_in,
                              void* d_out, int out_size, void* d_ws,
                              size_t ws_size, hipStream_t stream) {
  const float* h_in = (const float*)d_in[0];
  const int* src = (const int*)d_in[1];
  const int* dst = (const int*)d_in[2];
  const float* W1 = (const float*)d_in[3];
  const float* b1 = (const float*)d_in[4];
  const float* W2 = (const float*)d_in[5];
  const float* b2 = (const float*)d_in[6];
  const float* eps = (const float*)d_in[7];
  const float* gamma = (const float*)d_in[8];
  const float* beta = (const float*)d_in[9];
  float* out = (float*)d_out;

  const size_t rowBytes = (size_t)N_NODES * DIM * sizeof(float);
  const size_t wBytes = (size_t)N_LAYERS * DIM * DIM * sizeof(float);
  float* agg = (float*)d_ws;
  float* bufA = (float*)((char*)d_ws + rowBytes);
  float2* Wp1 = (float2*)((char*)d_ws + 2 * rowBytes);
  float2* Wp2 = (float2*)((char*)d_ws + 2 * rowBytes + wBytes);

  // Pack all 8 weight matrices into the WMMA B-fragment layout (L2-resident).
  {
    int total = N_LAYERS * (DIM / 4) * 2 * DIM;
    int blocks = (total + 255) / 256;
    pack_weights<<<blocks, 256, 0, stream>>>(W1, Wp1, N_LAYERS);
    pack_weights<<<blocks, 256, 0, stream>>>(W2, Wp2, N_LAYERS);
  }

  const int scatterBlocks = N_EDGES / 8;          // one wave32 per edge
  const int mlpBlocks = (N_NODES + TM - 1) / TM;  // 1563

  const float* cur = h_in;
  for (int layer = 0; layer < N_LAYERS; ++layer) {
    float* next = (layer % 2 == 0) ? bufA : out;  // layer 3 -> out
    hipMemsetAsync(agg, 0, rowBytes, stream);
    gin_scatter<<<scatterBlocks, 256, 0, stream>>>(cur, src, dst, agg);
    gin_mlp<<<mlpBlocks, 256, 0, stream>>>(
        cur, agg,
        (const v2f*)(Wp1 + (size_t)layer * (DIM / 4) * 2 * DIM),
        b1 + (size_t)layer * DIM,
        (const v2f*)(Wp2 + (size_t)layer * (DIM / 4) * 2 * DIM),
        b2 + (size_t)layer * DIM,
        eps, layer, gamma, beta, next);
    cur = next;
  }
}